// StokenAttention2_79817672229209
// MI455X (gfx1250) — compile-verified
//
#include <hip/hip_runtime.h>

typedef __attribute__((ext_vector_type(2))) float v2f;
typedef __attribute__((ext_vector_type(8))) float v8f;

#define HW 14400                         // 120*120
#define SCALE1_F 0.35355339059327373f    // 8^-0.5

// ---------------------------------------------------------------------------
// Depthwise conv: out[c, oy, ox] = b[c] + sum w[c,ky,kx] * x[c, oy*s-p+ky, ox*s-p+kx]
// x: 32x120x120; out rows have padded channel stride cs (>= hs*hs).
// ---------------------------------------------------------------------------
__global__ void dwconv_k(const float* __restrict__ x, const float* __restrict__ w,
                         const float* __restrict__ b, float* __restrict__ out,
                         int ksz, int stride, int pad, int hs, int cs) {
  int idx = blockIdx.x * blockDim.x + threadIdx.x;
  int total = 32 * hs * hs;
  if (idx >= total) return;
  int c = idx / (hs * hs);
  int rem = idx - c * hs * hs;
  int oy = rem / hs, ox = rem - oy * hs;
  int iy0 = oy * stride - pad;
  int ix0 = ox * stride - pad;
  const float* wc = w + c * ksz * ksz;
  const float* xc = x + c * HW;
  float acc = b[c];
  for (int ky = 0; ky < ksz; ++ky) {
    int iy = iy0 + ky;
    if (iy < 0 || iy >= 120) continue;
    for (int kx = 0; kx < ksz; ++kx) {
      int ix = ix0 + kx;
      if (ix < 0 || ix >= 120) continue;
      acc = fmaf(wc[ky * ksz + kx], xc[iy * 120 + ix], acc);
    }
  }
  out[c * cs + rem] = acc;
}

// ---------------------------------------------------------------------------
// GEMM via V_WMMA_F32_16X16X4_F32 (full fp32): OUT[M, Ns] = W[M,K] * IN[K, Ns] (+bias)
// Requirements (all satisfied by the call sites):
//   - Ns multiple of 16 (padded channel stride) => every n-tile full, B loads
//     and D stores unconditional, EXEC all-ones for WMMA.
//   - M % 8 == 0 => store guard (row < M) is identical for all 8 D rows of a
//     lane, so a single exec-masked region covers the whole epilogue.
//   - K in {8, 32}: templated so the k-loop fully unrolls (loads hoisted, then
//     K/4 back-to-back WMMAs).
// One wave (32 threads) per 16x16 tile.
// A layout: lane L supplies W[m0 + (L&15)][k + 2*(L>>4) + {0,1}] in a.{x,y}.
// B layout: lane L supplies IN[k + 2*(L>>4) + {0,1}][n0 + (L&15)] in b.{x,y}.
// D layout: acc[i] = D[m0 + 8*(L>>4) + i][n0 + (L&15)].
// ---------------------------------------------------------------------------
template <int K, bool HAS_BIAS>
__global__ void gemm_wmma_k(const float* __restrict__ W, const float* __restrict__ IN,
                            const float* __restrict__ bias, float* __restrict__ OUT,
                            int M, int Ns) {
  const int lane = threadIdx.x & 31;
  const int lm = lane & 15;
  const int lh = lane >> 4;
  const int n0 = blockIdx.x * 16;
  const int m0 = blockIdx.y * 16;

  const int mrow = m0 + lm;
  const bool mok = (mrow < M);
  const int wr = mok ? mrow : 0;      // clamped row -> loads always in-bounds
  const int ncol = n0 + lm;

  v2f a[K / 4], b[K / 4];
#pragma unroll
  for (int kk = 0; kk < K / 4; ++kk) {
    const int ka = kk * 4 + 2 * lh;   // this lane supplies K = ka, ka+1
    float w0 = W[wr * K + ka];        // unconditional loads
    float w1 = W[wr * K + ka + 1];
    a[kk].x = mok ? w0 : 0.f;         // value select (v_cndmask), no branching
    a[kk].y = mok ? w1 : 0.f;
    b[kk].x = IN[ka * Ns + ncol];
    b[kk].y = IN[(ka + 1) * Ns + ncol];
  }

  v8f acc = {};
#pragma unroll
  for (int kk = 0; kk < K / 4; ++kk) {
    acc = __builtin_amdgcn_wmma_f32_16x16x4_f32(
        /*neg_a=*/false, a[kk], /*neg_b=*/false, b[kk],
        /*c_mod=*/(short)0, acc, /*reuse_a=*/false, /*reuse_b=*/false);
  }

  const int rbase = m0 + 8 * lh;
  if (rbase < M) {                    // M % 8 == 0 -> covers all 8 rows below
#pragma unroll
    for (int i = 0; i < 8; ++i) {
      int row = rbase + i;
      float v = acc[i];
      if (HAS_BIAS) v += bias[row];
      OUT[row * Ns + ncol] = v;
    }
  }
}

// ---------------------------------------------------------------------------
// Attention with head_dim == 1 (channel stride cs, valid length N, N % 4 == 0):
//  q = qkv[3h], k = qkv[3h+1], v = qkv[3h+2]
//  out[h,m] = sum_n v[n]*exp(q[m]*k[n] - mx) / sum_n exp(q[m]*k[n] - mx)
//  mx = exact column max = q[m]>=0 ? q[m]*kmax : q[m]*kmin
// k,v interleaved in LDS: one ds_load_b64 per n feeds exp + 2 FMA.
// ---------------------------------------------------------------------------
__global__ void attn_k(const float* __restrict__ qkv, float* __restrict__ out,
                       int N, int cs) {
  const int h = blockIdx.y;
  const float* q = qkv + (3 * h + 0) * cs;
  const float* k = qkv + (3 * h + 1) * cs;
  const float* v = qkv + (3 * h + 2) * cs;

  __shared__ float2 skv[3600];
  __shared__ float sred[512];

  const int tid = threadIdx.x;
  float kmin = 3.4e38f, kmax = -3.4e38f;
  for (int i = tid; i < N; i += 256) {
    float kv = k[i];
    skv[i] = make_float2(kv, v[i]);
    kmin = fminf(kmin, kv);
    kmax = fmaxf(kmax, kv);
  }
  sred[tid] = kmin;
  sred[256 + tid] = kmax;
  __syncthreads();
  for (int s = 128; s > 0; s >>= 1) {
    if (tid < s) {
      sred[tid]       = fminf(sred[tid], sred[tid + s]);
      sred[256 + tid] = fmaxf(sred[256 + tid], sred[256 + tid + s]);
    }
    __syncthreads();
  }
  kmin = sred[0];
  kmax = sred[256];

  int m = blockIdx.x * 256 + tid;
  if (m < N) {
    float qm = q[m];
    float mx = (qm >= 0.f) ? qm * kmax : qm * kmin;
    float num = 0.f, den = 0.f;
    for (int n = 0; n < N; n += 4) {
#pragma unroll
      for (int u = 0; u < 4; ++u) {
        float2 t = skv[n + u];
        float e = __expf(fmaf(qm, t.x, -mx));
        num = fmaf(e, t.y, num);
        den += e;
      }
    }
    out[h * cs + m] = num / den;
  }
}

// ---------------------------------------------------------------------------
// scale_branch: thread per (superpixel n, sub-position p). Fully independent.
// sf/sr have padded channel stride cs.
// ---------------------------------------------------------------------------
__global__ void scale_branch_k(const float* __restrict__ xp, const float* __restrict__ sf,
                               const float* __restrict__ sr, float* __restrict__ out,
                               int hs, int r, int cs) {
  const int N = hs * hs;
  const int P = r * r;
  int idx = blockIdx.x * blockDim.x + threadIdx.x;
  if (idx >= N * P) return;
  int n = idx / P, p = idx - n * P;
  int row = n / hs, col = n - row * hs;
  int pr = p / r, pc = p - pr * r;
  int Y = row * r + pr, X = col * r + pc;

  float xg[8];
#pragma unroll
  for (int c = 0; c < 8; ++c) xg[c] = xp[c * HW + Y * 120 + X];

  float aff[9];
  int nb[9];
#pragma unroll
  for (int q = 0; q < 9; ++q) {
    int i = q / 3, j = q - i * 3;
    int rr = row + i - 1, cc = col + j - 1;
    bool ok = (rr >= 0) && (rr < hs) && (cc >= 0) && (cc < hs);
    int base = rr * hs + cc;
    nb[q] = ok ? base : -1;
    float dot = 0.f;
    if (ok) {
#pragma unroll
      for (int c = 0; c < 8; ++c) dot = fmaf(xg[c], sf[c * cs + base], dot);
    }
    aff[q] = dot * SCALE1_F;   // OOB patch -> zero values -> logit 0 (kept in softmax)
  }
  float mx = aff[0];
#pragma unroll
  for (int q = 1; q < 9; ++q) mx = fmaxf(mx, aff[q]);
  float s = 0.f;
#pragma unroll
  for (int q = 0; q < 9; ++q) { aff[q] = __expf(aff[q] - mx); s += aff[q]; }
  float inv = 1.f / s;

#pragma unroll
  for (int c = 0; c < 8; ++c) {
    float acc = 0.f;
#pragma unroll
    for (int q = 0; q < 9; ++q)
      if (nb[q] >= 0) acc = fmaf(sr[c * cs + nb[q]], aff[q], acc);
    out[c * HW + Y * 120 + X] = acc * inv;
  }
}

// ---------------------------------------------------------------------------
extern "C" void kernel_launch(void* const* d_in, const int* in_sizes, int n_in,
                              void* d_out, int out_size, void* d_ws, size_t ws_size,
                              hipStream_t stream) {
  const float* x      = (const float*)d_in[0];
  const float* C1_w   = (const float*)d_in[1];
  const float* C1_b   = (const float*)d_in[2];
  const float* C2_w   = (const float*)d_in[3];
  const float* C2_b   = (const float*)d_in[4];
  const float* C3_w   = (const float*)d_in[5];
  const float* C3_b   = (const float*)d_in[6];
  const float* pws_w  = (const float*)d_in[7];
  const float* pws_b  = (const float*)d_in[8];
  const float* qkv_w  = (const float*)d_in[9];
  const float* proj_w = (const float*)d_in[10];
  const float* proj_b = (const float*)d_in[11];
  float* out = (float*)d_out;
  float* ws  = (float*)d_ws;

  // Padded channel strides (multiples of 16): 3600, 912, 400, 14400
  // Workspace layout (floats); total ~393K floats (~1.6 MB)
  float* dw1  = ws;              // 32*3600 = 115200
  float* dw2  = dw1  + 115200;   // 32*912  = 29184
  float* dw3  = dw2  + 29184;    // 32*400  = 12800
  float* sf1  = dw3  + 12800;    // 8*3600  = 28800
  float* sf2  = sf1  + 28800;    // 8*912   = 7296
  float* sf3  = sf2  + 7296;     // 8*400   = 3200
  float* qkv1 = sf3  + 3200;     // 24*3600 = 86400
  float* qkv2 = qkv1 + 86400;    // 24*912  = 21888
  float* qkv3 = qkv2 + 21888;    // 24*400  = 9600
  float* ao1  = qkv3 + 9600;     // 8*3600  = 28800
  float* ao2  = ao1  + 28800;    // 8*912   = 7296
  float* ao3  = ao2  + 7296;     // 8*400   = 3200
  float* sr1  = ao3  + 3200;     // 8*3600  = 28800
  float* sr2  = sr1  + 28800;    // 8*912   = 7296
  float* sr3  = sr2  + 7296;     // 8*400   = 3200
  float* xp   = out + 24 * HW;   // xp written straight into output channels 24..31

  // 1) depthwise convs (write with padded channel stride)
  dwconv_k<<<dim3(450), 256, 0, stream>>>(x, C1_w, C1_b, dw1,  4, 2, 1, 60, 3600);
  dwconv_k<<<dim3(113), 256, 0, stream>>>(x, C2_w, C2_b, dw2,  8, 4, 2, 30, 912);
  dwconv_k<<<dim3(50),  256, 0, stream>>>(x, C3_w, C3_b, dw3, 12, 6, 3, 20, 400);

  // 2) pointwise 8x32 conv (WMMA fp32): sf_i and xp
  gemm_wmma_k<32, true><<<dim3(225, 1), 32, 0, stream>>>(pws_w, dw1, pws_b, sf1, 8, 3600);
  gemm_wmma_k<32, true><<<dim3(57,  1), 32, 0, stream>>>(pws_w, dw2, pws_b, sf2, 8, 912);
  gemm_wmma_k<32, true><<<dim3(25,  1), 32, 0, stream>>>(pws_w, dw3, pws_b, sf3, 8, 400);
  gemm_wmma_k<32, true><<<dim3(900, 1), 32, 0, stream>>>(pws_w, x,   pws_b, xp,  8, HW);

  // 3) qkv = qkv_w(24x8) @ sf_i (no bias)
  gemm_wmma_k<8, false><<<dim3(225, 2), 32, 0, stream>>>(qkv_w, sf1, nullptr, qkv1, 24, 3600);
  gemm_wmma_k<8, false><<<dim3(57,  2), 32, 0, stream>>>(qkv_w, sf2, nullptr, qkv2, 24, 912);
  gemm_wmma_k<8, false><<<dim3(25,  2), 32, 0, stream>>>(qkv_w, sf3, nullptr, qkv3, 24, 400);

  // 4) head-dim-1 attention (exact streaming softmax over n)
  attn_k<<<dim3(15, 8), 256, 0, stream>>>(qkv1, ao1, 3600, 3600);
  attn_k<<<dim3(4,  8), 256, 0, stream>>>(qkv2, ao2, 900,  912);
  attn_k<<<dim3(2,  8), 256, 0, stream>>>(qkv3, ao3, 400,  400);

  // 5) proj = proj_w(8x8) @ ao_i + proj_b
  gemm_wmma_k<8, true><<<dim3(225, 1), 32, 0, stream>>>(proj_w, ao1, proj_b, sr1, 8, 3600);
  gemm_wmma_k<8, true><<<dim3(57,  1), 32, 0, stream>>>(proj_w, ao2, proj_b, sr2, 8, 912);
  gemm_wmma_k<8, true><<<dim3(25,  1), 32, 0, stream>>>(proj_w, ao3, proj_b, sr3, 8, 400);

  // 6) scale branches -> output channels 0..23 (xp already in 24..31)
  scale_branch_k<<<dim3(57), 256, 0, stream>>>(xp, sf1, sr1, out,           60, 2, 3600);
  scale_branch_k<<<dim3(57), 256, 0, stream>>>(xp, sf2, sr2, out + 8 * HW,  30, 4, 912);
  scale_branch_k<<<dim3(57), 256, 0, stream>>>(xp, sf3, sr3, out + 16 * HW, 20, 6, 400);
}